// RankingLoss_35862976921747
// MI455X (gfx1250) — compile-verified
//
#include <hip/hip_runtime.h>
#include <hip/hip_bf16.h>
#include <cstddef>

#define NN 8192
#define TOL 0.01f

typedef __attribute__((ext_vector_type(2))) float v2f;
typedef __attribute__((ext_vector_type(8))) float v8f;

// ---------------------------------------------------------------------------
// Kernel 1: rank-by-counting "argsort(-labels)" gather.
// rank[k] = #{m : labels[m] > labels[k]  ||  (labels[m]==labels[k] && m<k)}
// (matches stable ascending sort of -labels). Then s_sorted[rank] = logits[k].
// 64M compares total -- negligible next to the 256MB output pass.
// ---------------------------------------------------------------------------
__global__ void rank_sort_kernel(const float* __restrict__ logits,
                                 const float* __restrict__ labels,
                                 float* __restrict__ s_sorted, int n) {
    __shared__ float tile[256];
    const int k = blockIdx.x * 256 + threadIdx.x;
    const float lk = labels[k];
    int rank = 0;
    for (int base = 0; base < n; base += 256) {
        tile[threadIdx.x] = labels[base + threadIdx.x];
        __syncthreads();
        #pragma unroll 8
        for (int t = 0; t < 256; ++t) {
            const float lm = tile[t];
            const int m = base + t;
            rank += ((lm > lk) || (lm == lk && m < k)) ? 1 : 0;
        }
        __syncthreads();
    }
    s_sorted[rank] = logits[k];
}

// ---------------------------------------------------------------------------
// Kernel 2: one 16x16 output tile per wave via V_WMMA_F32_16X16X4_F32.
// diff[i][j] = s[i] - s[j] built as rank-2 outer product:
//   A(16x4) columns = [ s_i , 1 , 0 , 0 ],  B(4x16) rows = [ 1 ; -s_j ; 0 ; 0 ]
// A layout (2 VGPRs): lanes 0-15 -> K=0,1 ; lanes 16-31 -> K=2,3 (zeroed).
// C/D layout: VGPR r, lanes 0-15 -> (M=r, N=lane); lanes 16-31 -> (M=r+8).
// Epilogue: out = (j>i && |d|>TOL) ? min(d,0)-log(1+exp(-|d|)) : 0.
// Tiles strictly below the diagonal (tj<ti) are all-zero: store-only path.
// ---------------------------------------------------------------------------
__global__ void ranking_loss_tile_kernel(const float* __restrict__ s,
                                         float* __restrict__ out) {
    const int lane = threadIdx.x & 31;
    const int wave = threadIdx.x >> 5;
    const int ti = blockIdx.x;                 // tile row   0..511
    const int tj = blockIdx.y * 8 + wave;      // tile col   0..511
    const int col = lane & 15;
    const bool lo = (lane < 16);

    const int i0 = ti * 16;
    const int j = tj * 16 + col;
    const int rbase = i0 + (lo ? 0 : 8);

    if (tj < ti) {
        // whole tile is j<i -> zeros (half the matrix): store-only fast path
        #pragma unroll
        for (int r = 0; r < 8; ++r)
            out[(size_t)(rbase + r) * NN + j] = 0.0f;
        return;
    }

    const float si = s[i0 + col];   // 16-float broadcast, L2-resident (32KB)
    const float sj = s[j];

    v2f a, b;
    a.x = lo ?  si   : 0.0f;   // A[:,0] = s_i        (K=2 lanes carry zeros)
    a.y = lo ?  1.0f : 0.0f;   // A[:,1] = 1          (K=3 lanes carry zeros)
    b.x = lo ?  1.0f : 0.0f;   // B[0,:] = 1
    b.y = lo ? -sj   : 0.0f;   // B[1,:] = -s_j
    v8f c = {};
    // (neg_a, A, neg_b, B, c_mod, C, reuse_a, reuse_b)
    c = __builtin_amdgcn_wmma_f32_16x16x4_f32(false, a, false, b,
                                              (short)0, c, false, false);

    #pragma unroll
    for (int r = 0; r < 8; ++r) {
        const int i = rbase + r;
        const float d = c[r];
        const float ad = fabsf(d);
        // log_sigmoid(d) = min(d,0) - log1p(exp(-|d|))
        const float ls = fminf(d, 0.0f) - __logf(1.0f + __expf(-ad));
        const float o = ((j > i) && (ad > TOL)) ? ls : 0.0f;
        out[(size_t)i * NN + j] = o;   // lanes 0-15 / 16-31: two 64B segments
    }
}

extern "C" void kernel_launch(void* const* d_in, const int* in_sizes, int n_in,
                              void* d_out, int out_size, void* d_ws, size_t ws_size,
                              hipStream_t stream) {
    const float* logits = (const float*)d_in[0];
    const float* labels = (const float*)d_in[1];
    float* out = (float*)d_out;
    float* s_sorted = (float*)d_ws;            // 8192 floats = 32KB scratch
    const int n = in_sizes[0];                 // 8192

    rank_sort_kernel<<<n / 256, 256, 0, stream>>>(logits, labels, s_sorted, n);

    const int tiles = n / 16;                  // 512
    dim3 grid(tiles, tiles / 8);               // 8 waves/block -> 8 tile-cols
    ranking_loss_tile_kernel<<<grid, 256, 0, stream>>>(s_sorted, out);
}